// GAT_26903675142174
// MI455X (gfx1250) — compile-verified
//
#include <hip/hip_runtime.h>
#include <hip/hip_bf16.h>
#include <math.h>

typedef float v2f __attribute__((ext_vector_type(2)));
typedef float v8f __attribute__((ext_vector_type(8)));

#define NEG_SLOPE 0.2f
#define EPSV      1e-16f

// ---------------------------------------------------------------------------
// FP32 GEMM via V_WMMA_F32_16X16X4_F32. One wave32 per 16x16 C tile.
// A: [M,K] row-major, B: [K,N] row-major, C: [M,N] row-major.
// Requires M%16==0, N%16==0, K%4==0 (true for this problem: M=50000, K=128).
// ---------------------------------------------------------------------------
__global__ void gemm_wmma_f32(const float* __restrict__ A,
                              const float* __restrict__ B,
                              float* __restrict__ C,
                              int M, int N, int K) {
  int gtid = blockIdx.x * blockDim.x + threadIdx.x;
  int wave = gtid >> 5;
  int lane = threadIdx.x & 31;
  int ntiles = N >> 4;
  int mt = wave / ntiles;
  int nt = wave - mt * ntiles;
  if (mt * 16 >= M) return;             // whole wave exits together

  const int mrow = mt * 16 + (lane & 15);     // A row this lane feeds
  const int ncol = nt * 16 + (lane & 15);     // B col this lane feeds
  const int kb   = (lane >> 4) << 1;          // lanes 0-15: K=0,1; 16-31: K=2,3

  v8f acc = {0.f, 0.f, 0.f, 0.f, 0.f, 0.f, 0.f, 0.f};
  const float* Ap = A + (size_t)mrow * K + kb;
  const float* Bp = B + (size_t)kb * N + ncol;

#pragma unroll 4
  for (int k = 0; k < K; k += 4) {
    v2f a = *(const v2f*)(Ap + k);            // A[mrow][k+kb], A[mrow][k+kb+1]
    v2f b;
    b.x = Bp[(size_t)k * N];                  // B[k+kb][ncol]
    b.y = Bp[(size_t)(k + 1) * N];            // B[k+kb+1][ncol]
    acc = __builtin_amdgcn_wmma_f32_16x16x4_f32(
        /*neg_a=*/false, a, /*neg_b=*/false, b,
        /*c_mod=*/(short)0, acc, /*reuse_a=*/false, /*reuse_b=*/false);
  }

  // D layout: VGPR r -> row (mt*16 + r + (lane>=16 ? 8 : 0)), col nt*16+(lane&15)
  int row0 = mt * 16 + ((lane >> 4) << 3);
  float* Cp = C + (size_t)row0 * N + ncol;
#pragma unroll
  for (int r = 0; r < 8; ++r) Cp[(size_t)r * N] = acc[r];
}

// ---------------------------------------------------------------------------
// Per-(node, head) attention coefficients: a_src/a_dst = <h[n,h,:], att[h,:]>
// ---------------------------------------------------------------------------
__global__ void attn_coeff(const float* __restrict__ h,
                           const float* __restrict__ att_s,
                           const float* __restrict__ att_d,
                           float* __restrict__ a_s, float* __restrict__ a_d,
                           int N, int H, int C) {
  int t = blockIdx.x * blockDim.x + threadIdx.x;   // t = n*H + head
  if (t >= N * H) return;
  int hd = t % H;
  const float* hp = h + (size_t)t * C;             // [N,H,C] contiguous
  const float* as = att_s + (size_t)hd * C;
  const float* ad = att_d + (size_t)hd * C;
  float ss = 0.f, sd = 0.f;
  for (int c = 0; c < C; ++c) {
    float v = hp[c];
    ss += v * as[c];
    sd += v * ad[c];
  }
  a_s[t] = ss;
  a_d[t] = sd;
}

// ordered-int encoding so integer atomicMax implements float max
__device__ __forceinline__ int f2oi(float f) {
  int i = __float_as_int(f);
  return i >= 0 ? i : (i ^ 0x7FFFFFFF);
}
__device__ __forceinline__ float oi2f(int i) {
  return __int_as_float(i >= 0 ? i : (i ^ 0x7FFFFFFF));
}

__device__ __forceinline__ void edge_nodes(const int* __restrict__ ei, int E,
                                           int e, int& src, int& dst) {
  if (e < E) { src = ei[e]; dst = ei[E + e]; }
  else       { src = dst = e - E; }              // appended self loops
}

__global__ void fill_i32(int* p, int v, long long n) {
  long long t = (long long)blockIdx.x * blockDim.x + threadIdx.x;
  if (t < n) p[t] = v;
}
__global__ void fill_f32(float* p, float v, long long n) {
  long long t = (long long)blockIdx.x * blockDim.x + threadIdx.x;
  if (t < n) p[t] = v;
}

// pass 1: segment max of leaky-relu logits into m[dst,head] (ordered-int)
__global__ void edge_max_k(const int* __restrict__ ei, int E, int N, int H,
                           const float* __restrict__ a_s,
                           const float* __restrict__ a_d,
                           int* __restrict__ m) {
  long long t = (long long)blockIdx.x * blockDim.x + threadIdx.x;
  long long total = (long long)(E + N) * H;
  if (t >= total) return;
  int e = (int)(t / H), hd = (int)(t % H);
  int src, dst; edge_nodes(ei, E, e, src, dst);
  float lg = a_s[(size_t)src * H + hd] + a_d[(size_t)dst * H + hd];
  lg = lg > 0.f ? lg : lg * NEG_SLOPE;
  atomicMax(&m[(size_t)dst * H + hd], f2oi(lg));
}

// pass 2: e = exp(logit - m[dst]); store into alpha; segment-sum into s[dst]
__global__ void edge_exp_k(const int* __restrict__ ei, int E, int N, int H,
                           const float* __restrict__ a_s,
                           const float* __restrict__ a_d,
                           const int* __restrict__ m,
                           float* __restrict__ alpha, float* __restrict__ s) {
  long long t = (long long)blockIdx.x * blockDim.x + threadIdx.x;
  long long total = (long long)(E + N) * H;
  if (t >= total) return;
  int e = (int)(t / H), hd = (int)(t % H);
  int src, dst; edge_nodes(ei, E, e, src, dst);
  float lg = a_s[(size_t)src * H + hd] + a_d[(size_t)dst * H + hd];
  lg = lg > 0.f ? lg : lg * NEG_SLOPE;
  float ev = expf(lg - oi2f(m[(size_t)dst * H + hd]));
  alpha[t] = ev;
  atomicAdd(&s[(size_t)dst * H + hd], ev);
}

// pass 3: alpha /= (s[dst] + eps)
__global__ void edge_div_k(const int* __restrict__ ei, int E, int N, int H,
                           const float* __restrict__ s, float* __restrict__ alpha) {
  long long t = (long long)blockIdx.x * blockDim.x + threadIdx.x;
  long long total = (long long)(E + N) * H;
  if (t >= total) return;
  int e = (int)(t / H), hd = (int)(t % H);
  int src, dst; edge_nodes(ei, E, e, src, dst);
  alpha[t] = alpha[t] / (s[(size_t)dst * H + hd] + EPSV);
}

// pass 4: out[dst, f] += h[src, f] * alpha[e, f/C]   (float4 per thread)
__global__ void edge_scatter_k(const int* __restrict__ ei, int E, int N,
                               int H, int C,
                               const float* __restrict__ h,
                               const float* __restrict__ alpha,
                               float* __restrict__ out) {
  int F = H * C;
  int quads = F >> 2;
  long long t = (long long)blockIdx.x * blockDim.x + threadIdx.x;
  long long total = (long long)(E + N) * quads;
  if (t >= total) return;
  int e = (int)(t / quads);
  int q = (int)(t - (long long)e * quads);
  int f0 = q << 2;
  int hd = f0 / C;                                 // 4 consecutive f share a head
  int src, dst; edge_nodes(ei, E, e, src, dst);
  float al = alpha[(size_t)e * H + hd];
  float4 v = *(const float4*)(h + (size_t)src * F + f0);
  float* op = out + (size_t)dst * F + f0;
  atomicAdd(op + 0, v.x * al);
  atomicAdd(op + 1, v.y * al);
  atomicAdd(op + 2, v.z * al);
  atomicAdd(op + 3, v.w * al);
}

// layer-1 epilogue: out = elu(out + bias)  (in place, bias over F=H*C)
__global__ void bias_elu_k(float* __restrict__ o, const float* __restrict__ bias,
                           long long n, int F) {
  long long t = (long long)blockIdx.x * blockDim.x + threadIdx.x;
  if (t >= n) return;
  int f = (int)(t % F);
  float v = o[t] + bias[f];
  o[t] = v > 0.f ? v : (expf(v) - 1.f);
}

// layer-2 epilogue: mean over heads + bias2
__global__ void final_mean_bias_k(const float* __restrict__ acc,
                                  const float* __restrict__ bias,
                                  float* __restrict__ out, int N, int H, int C) {
  int t = blockIdx.x * blockDim.x + threadIdx.x;   // t = n*C + c
  if (t >= N * C) return;
  int n = t / C, c = t - n * C;
  const float* p = acc + (size_t)n * H * C + c;
  float s = 0.f;
  for (int h = 0; h < H; ++h) s += p[(size_t)h * C];
  out[t] = s * (1.0f / H) + bias[c];
}

static inline int nblk(long long n, int b) { return (int)((n + b - 1) / b); }

extern "C" void kernel_launch(void* const* d_in, const int* in_sizes, int n_in,
                              void* d_out, int out_size, void* d_ws, size_t ws_size,
                              hipStream_t stream) {
  const float* x        = (const float*)d_in[0];
  const int*   ei       = (const int*)d_in[1];
  const float* W1       = (const float*)d_in[2];
  const float* att_src1 = (const float*)d_in[3];
  const float* att_dst1 = (const float*)d_in[4];
  const float* bias1    = (const float*)d_in[5];
  const float* W2       = (const float*)d_in[6];
  const float* att_src2 = (const float*)d_in[7];
  const float* att_dst2 = (const float*)d_in[8];
  const float* bias2    = (const float*)d_in[9];

  const int Fin = 128, H = 4, C1 = 32, C2 = 64;
  const int F1 = H * C1;                 // 128
  const int F2 = H * C2;                 // 256
  const int N  = in_sizes[0] / Fin;      // 50000
  const int E  = in_sizes[1] / 2;        // 800000
  const int Et = E + N;                  // edges incl. self loops
  const int B  = 256;
  const int NEG_INF_OI = (int)0x807FFFFF; // f2oi(-inf)

  // ---- workspace carve-up (256B aligned regions) ----
  char* w = (char*)d_ws;
  auto carve = [&](size_t bytes) -> void* {
    void* p = (void*)w;
    w += (bytes + 255) & ~(size_t)255;
    return p;
  };
  float* h1    = (float*)carve((size_t)N * F1 * 4);   // x @ W1          [N,128]
  float* o1    = (float*)carve((size_t)N * F1 * 4);   // layer1 out/elu  [N,128]
  float* h2    = (float*)carve((size_t)N * F2 * 4);   // o1 @ W2         [N,256]
  float* o2    = (float*)carve((size_t)N * F2 * 4);   // layer2 agg      [N,256]
  float* a_s   = (float*)carve((size_t)N * H * 4);
  float* a_d   = (float*)carve((size_t)N * H * 4);
  int*   m     = (int*)  carve((size_t)N * H * 4);
  float* s     = (float*)carve((size_t)N * H * 4);
  float* alpha = (float*)carve((size_t)Et * H * 4);

  const long long nhT  = (long long)N * H;      // node-head threads
  const long long ehT  = (long long)Et * H;     // edge-head threads

  // =========================== Layer 1 ===========================
  {
    long long waves = (long long)(N / 16) * (F1 / 16);
    gemm_wmma_f32<<<nblk(waves * 32, B), B, 0, stream>>>(x, W1, h1, N, F1, Fin);
    attn_coeff<<<nblk(nhT, B), B, 0, stream>>>(h1, att_src1, att_dst1, a_s, a_d, N, H, C1);

    fill_i32<<<nblk(nhT, B), B, 0, stream>>>(m, NEG_INF_OI, nhT);
    edge_max_k<<<nblk(ehT, B), B, 0, stream>>>(ei, E, N, H, a_s, a_d, m);
    fill_f32<<<nblk(nhT, B), B, 0, stream>>>(s, 0.f, nhT);
    edge_exp_k<<<nblk(ehT, B), B, 0, stream>>>(ei, E, N, H, a_s, a_d, m, alpha, s);
    edge_div_k<<<nblk(ehT, B), B, 0, stream>>>(ei, E, N, H, s, alpha);

    long long oN = (long long)N * F1;
    fill_f32<<<nblk(oN, B), B, 0, stream>>>(o1, 0.f, oN);
    long long sc = (long long)Et * (F1 / 4);
    edge_scatter_k<<<nblk(sc, B), B, 0, stream>>>(ei, E, N, H, C1, h1, alpha, o1);
    bias_elu_k<<<nblk(oN, B), B, 0, stream>>>(o1, bias1, oN, F1);
  }

  // =========================== Layer 2 ===========================
  {
    long long waves = (long long)(N / 16) * (F2 / 16);
    gemm_wmma_f32<<<nblk(waves * 32, B), B, 0, stream>>>(o1, W2, h2, N, F2, F1);
    attn_coeff<<<nblk(nhT, B), B, 0, stream>>>(h2, att_src2, att_dst2, a_s, a_d, N, H, C2);

    fill_i32<<<nblk(nhT, B), B, 0, stream>>>(m, NEG_INF_OI, nhT);
    edge_max_k<<<nblk(ehT, B), B, 0, stream>>>(ei, E, N, H, a_s, a_d, m);
    fill_f32<<<nblk(nhT, B), B, 0, stream>>>(s, 0.f, nhT);
    edge_exp_k<<<nblk(ehT, B), B, 0, stream>>>(ei, E, N, H, a_s, a_d, m, alpha, s);
    edge_div_k<<<nblk(ehT, B), B, 0, stream>>>(ei, E, N, H, s, alpha);

    long long oN = (long long)N * F2;
    fill_f32<<<nblk(oN, B), B, 0, stream>>>(o2, 0.f, oN);
    long long sc = (long long)Et * (F2 / 4);
    edge_scatter_k<<<nblk(sc, B), B, 0, stream>>>(ei, E, N, H, C2, h2, alpha, o2);

    final_mean_bias_k<<<nblk((long long)N * C2, B), B, 0, stream>>>(
        o2, bias2, (float*)d_out, N, H, C2);
  }
}